// FusedAttentionV3_14980845928806
// MI455X (gfx1250) — compile-verified
//
#include <hip/hip_runtime.h>
#include <math.h>

// ---------------------------------------------------------------------------
// Fused multi-head attention forward for MI455X (gfx1250), fp32 end to end.
//   1) QKV projection GEMM (WMMA f32 16x16x4), async-LDS double buffered,
//      bias fused, scatter to Q/K/V in [B,H,S,D] layout in workspace.
//   2) Flash-attention per (b,h): scores via K@Q^T (transposed so the online
//      softmax reduces mostly in-register), P@V via V^T@P^T, all WMMA f32.
//      K/V chunks streamed with GLOBAL_LOAD_ASYNC_TO_LDS, double buffered.
//   3) Output projection GEMM (same WMMA kernel), bias fused, writes d_out.
// ---------------------------------------------------------------------------

typedef float v2f __attribute__((ext_vector_type(2)));
typedef float v8f __attribute__((ext_vector_type(8)));

// D = A(16x4 f32) * B(4x16 f32) + C(16x16 f32)
#define WMMA_F32(a, b, c) \
  __builtin_amdgcn_wmma_f32_16x16x4_f32(false, (a), false, (b), (short)0, (c), false, false)

constexpr int Bn = 2, Sn = 2048, En = 1024, Hn = 16, Dn = 64;
constexpr int F3 = 3 * En;     // 3072
constexpr int Mtok = Bn * Sn;  // 4096

// Low 32 bits of a generic pointer to LDS == LDS byte offset (aperture form).
__device__ __forceinline__ uint32_t lds_addr32(const void* p) {
  return (uint32_t)(uintptr_t)p;
}
// Async DMA: 16B per lane, global -> LDS, no VGPR data round trip (ASYNCcnt).
__device__ __forceinline__ void async_b128(uint32_t lds, const float* g) {
  asm volatile("global_load_async_to_lds_b128 %0, %1, off"
               :: "v"(lds), "v"(g) : "memory");
}
__device__ __forceinline__ void wait_async0() {
  asm volatile("s_wait_asynccnt 0" ::: "memory");
}
// Same-wave LDS write->read ordering (DS in-order per wave; clobber stops
// the compiler reordering around the cross-lane LDS dependency).
__device__ __forceinline__ void lds_wave_fence() {
  asm volatile("s_wait_dscnt 0" ::: "memory");
}

// ---------------------------------------------------------------------------
// GEMM: C[m,n] = sum_k A[m,k] * W[n,k] + bias[n]
// Block tile 128(M) x 64(N), K in 32-wide chunks, double-buffered async LDS.
// 8 waves in a 4x2 grid; each wave owns a 32x32 C tile (4 WMMA accumulators)
// -> 32 WMMA per chunk against 32 ds_load_b64 (2x operand reuse).
// mode 0: scatter epilogue into Q/K/V [B,H,S,D].  mode 1: plain store to out.
// ---------------------------------------------------------------------------
__global__ __launch_bounds__(256) void gemm_wmma(
    const float* __restrict__ A, const float* __restrict__ W,
    const float* __restrict__ bias, float* __restrict__ outp,
    float* __restrict__ qp, float* __restrict__ kp, float* __restrict__ vp,
    int N, int K, int mode) {
  // stride 36 floats: rows 16B-aligned for async b128, banks spread for b64
  __shared__ float xs[2][128][36];  // A tiles (double buffered)
  __shared__ float bs[2][64][36];   // W tiles

  const int tid = threadIdx.x;
  const int w = tid >> 5, lane = tid & 31;
  const int l16 = lane & 15, hi = lane >> 4;
  const int wm = w & 3, wn = w >> 2;
  const int m0 = blockIdx.y * 128, n0 = blockIdx.x * 64;
  const int koffb = hi ? 2 : 0;  // A/B fragment K-offset per half-wave

  v8f acc[2][2] = {{v8f{}, v8f{}}, {v8f{}, v8f{}}};

  // issue one K-chunk of async global->LDS DMA (6 b128 per thread)
  auto issue = [&](int k0, int buf) {
#pragma unroll
    for (int j = 0; j < 4; ++j) {          // 128x32 A tile: 1024 float4
      const int fid = tid + j * 256;
      const int row = fid >> 3, c4 = (fid & 7) * 4;
      async_b128(lds_addr32(&xs[buf][row][c4]),
                 A + (size_t)(m0 + row) * K + k0 + c4);
    }
#pragma unroll
    for (int j = 0; j < 2; ++j) {          // 64x32 W tile: 512 float4
      const int fid = tid + j * 256;
      const int row = fid >> 3, c4 = (fid & 7) * 4;
      async_b128(lds_addr32(&bs[buf][row][c4]),
                 W + (size_t)(n0 + row) * K + k0 + c4);
    }
  };

  issue(0, 0);
  const int nch = K / 32;
  for (int c = 0; c < nch; ++c) {
    const int buf = c & 1;
    wait_async0();     // this wave's chunk-c DMA has landed
    __syncthreads();   // everyone's chunk-c landed; buf^1 reads all done
    if (c + 1 < nch) issue((c + 1) * 32, buf ^ 1);  // overlap with compute

#pragma unroll
    for (int kk = 0; kk < 32; kk += 4) {
      v2f a0 = *(const v2f*)&xs[buf][wm * 32 + l16][kk + koffb];
      v2f a1 = *(const v2f*)&xs[buf][wm * 32 + 16 + l16][kk + koffb];
      v2f b0 = *(const v2f*)&bs[buf][wn * 32 + l16][kk + koffb];
      v2f b1 = *(const v2f*)&bs[buf][wn * 32 + 16 + l16][kk + koffb];
      acc[0][0] = WMMA_F32(a0, b0, acc[0][0]);
      acc[0][1] = WMMA_F32(a0, b1, acc[0][1]);
      acc[1][0] = WMMA_F32(a1, b0, acc[1][0]);
      acc[1][1] = WMMA_F32(a1, b1, acc[1][1]);
    }
  }

  // Epilogue. C layout: VGPR r, lane -> (M = r + 8*hi, N = lane&15)
  const float bv0 = bias[n0 + wn * 32 + l16];
  const float bv1 = bias[n0 + wn * 32 + 16 + l16];
#pragma unroll
  for (int ms = 0; ms < 2; ++ms) {
#pragma unroll
    for (int r = 0; r < 8; ++r) {
      const int mg = m0 + wm * 32 + ms * 16 + r + 8 * hi;  // token row
      const int b = mg >> 11, s = mg & (Sn - 1);
#pragma unroll
      for (int ns = 0; ns < 2; ++ns) {
        const int ng = n0 + wn * 32 + ns * 16 + l16;  // output feature
        const float val = acc[ms][ns][r] + (ns ? bv1 : bv0);
        if (mode == 0) {
          const int t = ng >> 10, rem = ng & 1023;
          const int h = rem >> 6, d = rem & 63;
          float* base = (t == 0) ? qp : ((t == 1) ? kp : vp);
          base[(((size_t)(b * Hn + h)) * Sn + s) * Dn + d] = val;
        } else {
          outp[(size_t)mg * N + ng] = val;
        }
      }
    }
  }
}

// ---------------------------------------------------------------------------
// Flash attention. Block = one (b,h) slice, 8 waves x 16 queries = 128 rows.
// Per 16-key chunk (double-buffered async K/V staging):
//   S^T(key,query) = K_chunk(16xD) @ Q^T(Dx16)   [16 WMMA]
//   online softmax over keys (8 in-register + one shfl_xor(16))
//   ctx^T(d,query) += V^T(Dx16keys) @ P^T(16x16) [16 WMMA]
// ---------------------------------------------------------------------------
__global__ __launch_bounds__(256) void flash_attn(
    const float* __restrict__ qg, const float* __restrict__ kg,
    const float* __restrict__ vg, float* __restrict__ ctx) {
  __shared__ float Qs[8][16][66];  // per-wave Q block [q][d], pre-scaled
  __shared__ float Ks[2][16][68];  // K chunk [key][d] (async, 16B rows)
  __shared__ float Vs[2][16][68];  // V chunk [key][d] (async, read transposed)
  __shared__ float Pb[8][16][18];  // per-wave P^T staging [q][key]

  const int tid = threadIdx.x;
  const int w = tid >> 5, lane = tid & 31;
  const int l16 = lane & 15, hi = lane >> 4;
  const int koffb = hi ? 2 : 0;
  const int bh = blockIdx.x;  // b*H + h
  const int b = bh >> 4, h = bh & 15;
  const int q0 = (blockIdx.y * 8 + w) * 16;

  // fold 1/sqrt(D) and log2(e) into Q so softmax uses exp2 directly
  const float sc = 0.125f * 1.44269504088896f;
  const float* qrow = qg + ((size_t)bh * Sn + q0) * Dn;
  for (int i = lane; i < 16 * 64; i += 32) {
    const int r = i >> 6, c = i & 63;
    Qs[w][r][c] = qrow[r * 64 + c] * sc;
  }

  // async staging of one 16x64 K chunk + V chunk (1 b128 per thread each)
  const int srow = tid >> 4, sc4 = (tid & 15) * 4;
  auto issue = [&](int kc, int buf) {
    const float* kb = kg + ((size_t)bh * Sn + kc * 16) * Dn;
    const float* vb = vg + ((size_t)bh * Sn + kc * 16) * Dn;
    async_b128(lds_addr32(&Ks[buf][srow][sc4]), kb + srow * 64 + sc4);
    async_b128(lds_addr32(&Vs[buf][srow][sc4]), vb + srow * 64 + sc4);
  };

  v8f acc[4] = {v8f{}, v8f{}, v8f{}, v8f{}};  // ctx^T: 64 d-rows x 16 queries
  float m_run = -3.0e38f, l_run = 0.0f;

  issue(0, 0);
  for (int kc = 0; kc < Sn / 16; ++kc) {
    const int buf = kc & 1;
    wait_async0();     // this wave's chunk-kc K/V landed
    __syncthreads();   // everyone's landed; buf^1 reads all done (also Qs)
    if (kc + 1 < Sn / 16) issue(kc + 1, buf ^ 1);

    // S^T = K @ Q^T : M=key, N=query, K-dim = D = 64
    v8f sacc = {};
#pragma unroll
    for (int kk = 0; kk < 64; kk += 4) {
      v2f a  = *(const v2f*)&Ks[buf][l16][kk + koffb];   // A[key][d..d+1]
      v2f bq = *(const v2f*)&Qs[w][l16][kk + koffb];     // B[d][q] = Q[q][d]
      sacc = WMMA_F32(a, bq, sacc);
    }

    // online softmax over keys: lane holds 8 keys (r + 8*hi) for query l16
    float cmax = sacc[0];
#pragma unroll
    for (int r = 1; r < 8; ++r) cmax = fmaxf(cmax, sacc[r]);
    cmax = fmaxf(cmax, __shfl_xor(cmax, 16, 32));
    const float m_new = fmaxf(m_run, cmax);
    const float alpha = exp2f(m_run - m_new);
    float p[8];
    float psum = 0.0f;
#pragma unroll
    for (int r = 0; r < 8; ++r) { p[r] = exp2f(sacc[r] - m_new); psum += p[r]; }
    psum += __shfl_xor(psum, 16, 32);
    l_run = l_run * alpha + psum;
    m_run = m_new;

    // stage P^T as [q][key] so B-fragments are contiguous b64 loads
#pragma unroll
    for (int r = 0; r < 8; ++r) Pb[w][l16][r + 8 * hi] = p[r];
    lds_wave_fence();  // same-wave cross-lane LDS dependency

#pragma unroll
    for (int t = 0; t < 4; ++t)
#pragma unroll
      for (int r = 0; r < 8; ++r) acc[t][r] *= alpha;

    // ctx^T += V^T @ P^T : M = d (4 tiles of 16), K-dim = 16 keys
#pragma unroll
    for (int t = 0; t < 4; ++t) {
      const int d = t * 16 + l16;
#pragma unroll
      for (int s4 = 0; s4 < 4; ++s4) {
        const int kf = s4 * 4 + koffb;
        v2f a;  // A[m=d][k..k+1] = V[k][d] pair (fuses to ds_load_2addr_b32)
        a.x = Vs[buf][kf][d];
        a.y = Vs[buf][kf + 1][d];
        v2f bp = *(const v2f*)&Pb[w][l16][kf];  // B[k][q] = P^T
        acc[t] = WMMA_F32(a, bp, acc[t]);
      }
    }
  }

  // normalize and write ctx in [B,S,H,D] (== [B,S,E]) for the out projection
  const float rinv = 1.0f / l_run;
#pragma unroll
  for (int t = 0; t < 4; ++t)
#pragma unroll
    for (int r = 0; r < 8; ++r) {
      const int d = t * 16 + r + 8 * hi;
      const int sq = q0 + l16;
      ctx[(((size_t)(b * Sn + sq)) * Hn + h) * Dn + d] = acc[t][r] * rinv;
    }
}

// ---------------------------------------------------------------------------
extern "C" void kernel_launch(void* const* d_in, const int* in_sizes, int n_in,
                              void* d_out, int out_size, void* d_ws,
                              size_t ws_size, hipStream_t stream) {
  const float* x     = (const float*)d_in[0];  // [B,S,E]
  const float* w_qkv = (const float*)d_in[1];  // [3E,E]
  const float* b_qkv = (const float*)d_in[2];  // [3E]
  const float* w_out = (const float*)d_in[3];  // [E,E]
  const float* b_out = (const float*)d_in[4];  // [E]
  float* out = (float*)d_out;                  // [B,S,E]

  float* ws = (float*)d_ws;
  const size_t per = (size_t)Bn * Hn * Sn * Dn;  // 4,194,304 floats
  float* qp = ws;            // Q [B,H,S,D]
  float* kp = ws + per;      // K [B,H,S,D]
  float* vp = ws + 2 * per;  // V [B,H,S,D]
  float* cx = ws + 3 * per;  // ctx [B,S,E]

  // 1) QKV projection + bias + scatter
  gemm_wmma<<<dim3(F3 / 64, Mtok / 128), 256, 0, stream>>>(
      x, w_qkv, b_qkv, nullptr, qp, kp, vp, F3, En, 0);

  // 2) flash attention: grid = (B*H, S / (16 queries * 8 waves))
  flash_attn<<<dim3(Bn * Hn, Sn / 128), 256, 0, stream>>>(qp, kp, vp, cx);

  // 3) output projection + bias
  gemm_wmma<<<dim3(En / 64, Mtok / 128), 256, 0, stream>>>(
      cx, w_out, b_out, out, nullptr, nullptr, nullptr, En, En, 1);
}